// Embed_61091614818942
// MI455X (gfx1250) — compile-verified
//
#include <hip/hip_runtime.h>
#include <hip/hip_bf16.h>

// Embedding gather: out[t, :] = W_E[tokens[t], :] * sqrt(D_MODEL)
//
// Roofline: ~64 MiB gathered reads + 64 MiB streaming writes ≈ 128 MiB
// -> ~5.5 us floor at 23.3 TB/s. Purely HBM-bound; WMMA inapplicable
// (one-hot matmul would multiply DRAM traffic by ~1000x).
//
// Strategy: 256-thread blocks (8 wave32 waves), 4 tokens per block.
//  - 4 consecutive int32 indices -> single scalar s_load_b128.
//  - 4 independent global_load_b128 in flight per wave (s_clause) before
//    any wait -> 4x memory-level parallelism vs one-row-per-block.
//  - Non-temporal b128 stores: output is write-once, keep the 192 MB L2
//    holding W_E rows (196 MB ~ fits) for repeated-token reuse.

typedef __attribute__((ext_vector_type(4))) float v4f;

template<int D_MODEL, int TPB>
__global__ __launch_bounds__(256) void
embed_gather_fixed(const int* __restrict__ tokens,
                   const float* __restrict__ W_E,
                   float* __restrict__ out,
                   float scale)
{
    constexpr int NVEC = D_MODEL / 4;       // float4s per row (256)
    const int t0 = blockIdx.x * TPB;        // first token of this block
    const int i  = threadIdx.x;             // float4 index within row

    // 4 consecutive indices -> compiler merges into one s_load_b128.
    int toks[TPB];
#pragma unroll
    for (int j = 0; j < TPB; ++j)
        toks[j] = tokens[t0 + j];

    // Issue all row loads back-to-back: 4 global_load_b128 in flight.
    v4f v[TPB];
#pragma unroll
    for (int j = 0; j < TPB; ++j)
        v[j] = *((const v4f*)(W_E + (size_t)toks[j] * D_MODEL) + i);

    // Scale and stream out with NT hint.
#pragma unroll
    for (int j = 0; j < TPB; ++j) {
        v4f s = v[j] * scale;
        __builtin_nontemporal_store(
            s, (v4f*)(out + (size_t)(t0 + j) * D_MODEL) + i);
    }
}

// Generic fallback for arbitrary d_model / n_tokens.
__global__ __launch_bounds__(256) void
embed_gather_generic(const int* __restrict__ tokens,
                     const float* __restrict__ W_E,
                     float* __restrict__ out,
                     int n_tokens, int d_model, float scale)
{
    const int t = blockIdx.x;
    if (t >= n_tokens) return;
    const int tok = tokens[t];
    const v4f* __restrict__ src = (const v4f*)(W_E + (size_t)tok * d_model);
    v4f* __restrict__ dst       = (v4f*)(out + (size_t)t * d_model);
    const int nvec = d_model >> 2;
    for (int i = threadIdx.x; i < nvec; i += blockDim.x) {
        v4f v = src[i] * scale;
        __builtin_nontemporal_store(v, &dst[i]);
    }
}

extern "C" void kernel_launch(void* const* d_in, const int* in_sizes, int n_in,
                              void* d_out, int out_size, void* d_ws, size_t ws_size,
                              hipStream_t stream) {
    const int*   tokens = (const int*)d_in[0];   // [8, 2048] int32
    const float* W_E    = (const float*)d_in[1]; // [50257, 1024] fp32
    float*       out    = (float*)d_out;         // [8, 2048, 1024] fp32

    const int n_tokens = in_sizes[0];            // 16384
    const int d_model  = out_size / n_tokens;    // 1024
    const float scale  = sqrtf((float)d_model);  // 32.0f

    constexpr int TPB = 4;                       // tokens per block
    if (d_model == 1024 && (n_tokens % TPB) == 0) {
        dim3 grid(n_tokens / TPB);               // 4096 blocks
        dim3 block(256);                         // = NVEC; 8 wave32 waves
        embed_gather_fixed<1024, TPB><<<grid, block, 0, stream>>>(
            tokens, W_E, out, scale);
    } else {
        dim3 grid(n_tokens);
        dim3 block(256);
        embed_gather_generic<<<grid, block, 0, stream>>>(
            tokens, W_E, out, n_tokens, d_model, scale);
    }
}